// MultiSimilarityLoss_7533372637498
// MI455X (gfx1250) — compile-verified
//
#include <hip/hip_runtime.h>
#include <math.h>

typedef __attribute__((ext_vector_type(2))) float v2f;
typedef __attribute__((ext_vector_type(4))) float v4f;
typedef __attribute__((ext_vector_type(8))) float v8f;

#define B_N   8192
#define D_K   128
#define NK    (D_K / 4)        // 32 k-steps of V_WMMA_F32_16X16X4_F32
#define WAVES 2                // waves per workgroup
#define NTHR  (WAVES * 32)
#define ROWS_PER_WG (WAVES * 16)
#define BC    64               // columns staged in LDS per block iteration
#define LDSS  132              // padded row stride (dwords): bank-conflict-free b64 reads

#define THRESH    0.5f
#define MARGIN    0.1f
#define SCALE_POS 2.0f
#define SCALE_NEG 40.0f
#define EPSV      1e-5f

static_assert(B_N % ROWS_PER_WG == 0, "rows");
static_assert(B_N % BC == 0, "cols");

__global__ __launch_bounds__(NTHR) void ms_main(const float* __restrict__ feats,
                                                const int*   __restrict__ labels,
                                                float*       __restrict__ partials)
{
    __shared__ float sf[BC * LDSS];
    __shared__ int   sl[BC];
    __shared__ float sred[WAVES];

    const int tid  = threadIdx.x;
    const int wave = tid >> 5;
    const int lane = tid & 31;
    const int m16  = lane & 15;
    const int half = lane >> 4;

    const int mbase = blockIdx.x * ROWS_PER_WG + wave * 16;
    const float INF = __builtin_inff();

    // ---- A fragments: 16 rows x 128 K, resident in VGPRs for both passes ----
    // A layout (16x4 f32): lane l -> row l%16; v0/v1 hold K = 4k + 2*(l/16) + {0,1}
    v2f a[NK];
    {
        const float* arow = feats + (size_t)(mbase + m16) * D_K + 2 * half;
        #pragma unroll
        for (int kk = 0; kk < NK; ++kk)
            a[kk] = *(const v2f*)(arow + kk * 4);
    }
    // C/D layout: vgpr v, lane l -> row M = 8*(l/16)+v, col N = l%16
    int rlab[8];
    #pragma unroll
    for (int v = 0; v < 8; ++v)
        rlab[v] = labels[mbase + half * 8 + v];

    float minp[8], maxn[8], psum[8], nsum[8];
    #pragma unroll
    for (int v = 0; v < 8; ++v) { minp[v] = INF; maxn[v] = -INF; psum[v] = 0.f; nsum[v] = 0.f; }

    for (int pass = 0; pass < 2; ++pass) {
        for (int blk = 0; blk < B_N / BC; ++blk) {
            __syncthreads();
            // ---- cooperative stage: 64 cols x 128 dims -> LDS (padded) ----
            {
                const float* src = feats + (size_t)blk * BC * D_K;
                #pragma unroll
                for (int i = 0; i < (BC * D_K) / (NTHR * 4); ++i) {
                    int e   = i * (NTHR * 4) + tid * 4;
                    v4f val = *(const v4f*)(src + e);
                    int col = e >> 7;       // / D_K
                    int dd  = e & (D_K - 1);
                    *(v4f*)(&sf[col * LDSS + dd]) = val;
                }
                if (tid < BC) sl[tid] = labels[blk * BC + tid];
            }
            __syncthreads();

            for (int t = 0; t < BC / 16; ++t) {
                const int cb = t * 16;
                v8f c = {};
                // B layout (4x16 f32): lane l -> col l%16; v0/v1 hold K = 4k + 2*(l/16) + {0,1}
                const float* bp = &sf[(cb + m16) * LDSS + 2 * half];
                #pragma unroll
                for (int kk = 0; kk < NK; ++kk) {
                    v2f b = *(const v2f*)(bp + kk * 4);
                    c = __builtin_amdgcn_wmma_f32_16x16x4_f32(
                            false, a[kk], false, b, (short)0, c, false, false);
                }
                const int clab = sl[cb + m16];
                if (pass == 0) {
                    #pragma unroll
                    for (int v = 0; v < 8; ++v) {
                        float s    = c[v];
                        bool  same = (rlab[v] == clab);
                        minp[v] = fminf(minp[v], (same && s < 1.0f - EPSV) ? s :  INF);
                        maxn[v] = fmaxf(maxn[v], (!same)                   ? s : -INF);
                    }
                } else {
                    #pragma unroll
                    for (int v = 0; v < 8; ++v) {
                        float s    = c[v];
                        bool  same = (rlab[v] == clab);
                        bool  pok  = same && (s < 1.0f - EPSV) && (s - MARGIN < maxn[v]);
                        bool  nok  = (!same) && (s + MARGIN > minp[v]);
                        psum[v] += __expf(pok ? (-SCALE_POS) * (s - THRESH) : -INF);
                        nsum[v] += __expf(nok ? ( SCALE_NEG) * (s - THRESH) : -INF);
                    }
                }
            }
        }
        if (pass == 0) {
            // reduce min/max across the 16 lanes of each half; result broadcast to all 16
            #pragma unroll
            for (int v = 0; v < 8; ++v) {
                float mp = minp[v], mn = maxn[v];
                #pragma unroll
                for (int d = 1; d < 16; d <<= 1) {
                    mp = fminf(mp, __shfl_xor(mp, d, 32));
                    mn = fmaxf(mn, __shfl_xor(mn, d, 32));
                }
                minp[v] = mp; maxn[v] = mn;
            }
        }
    }

    // ---- per-row loss, fixed-order reduction to a per-block partial ----
    float acc = 0.f;
    #pragma unroll
    for (int v = 0; v < 8; ++v) {
        float ps = psum[v], ns = nsum[v];
        #pragma unroll
        for (int d = 1; d < 16; d <<= 1) {
            ps += __shfl_xor(ps, d, 32);
            ns += __shfl_xor(ns, d, 32);
        }
        bool valid = (minp[v] < INF) && (maxn[v] > -INF) && (ps > 0.f) && (ns > 0.f);
        float rl = log1pf(ps) * (1.0f / SCALE_POS) + log1pf(ns) * (1.0f / SCALE_NEG);
        acc += valid ? rl : 0.f;
    }
    // each of the 16 lanes in a half holds identical acc; combine the two halves
    float wavetot = acc / 16.0f;   // avoid double count: every lane contributed same value
    // exact: take lane 0 of each half instead of averaging
    wavetot = ((m16 == 0) ? acc : 0.f);
    wavetot += __shfl_xor(wavetot, 16, 32);  // half0 rows + half1 rows, lanes 0&16 valid
    if (lane == 0) sred[wave] = wavetot;
    __syncthreads();
    if (tid == 0) {
        float t = 0.f;
        #pragma unroll
        for (int w = 0; w < WAVES; ++w) t += sred[w];
        partials[blockIdx.x] = t;
    }
}

__global__ void ms_final(const float* __restrict__ partials, float* __restrict__ out)
{
    if (threadIdx.x == 0 && blockIdx.x == 0) {
        float s = 0.f;
        for (int i = 0; i < B_N / ROWS_PER_WG; ++i) s += partials[i];
        out[0] = s * (1.0f / (float)B_N);
    }
}

extern "C" void kernel_launch(void* const* d_in, const int* in_sizes, int n_in,
                              void* d_out, int out_size, void* d_ws, size_t ws_size,
                              hipStream_t stream) {
    (void)in_sizes; (void)n_in; (void)out_size; (void)ws_size;
    const float* feats  = (const float*)d_in[0];
    const int*   labels = (const int*)d_in[1];
    float* partials = (float*)d_ws;   // B_N/ROWS_PER_WG = 256 floats
    ms_main<<<dim3(B_N / ROWS_PER_WG), dim3(NTHR), 0, stream>>>(feats, labels, partials);
    ms_final<<<dim3(1), dim3(32), 0, stream>>>(partials, (float*)d_out);
}